// DGCNN_Grouper_37151467111132
// MI455X (gfx1250) — compile-verified
//
#include <hip/hip_runtime.h>
#include <hip/hip_bf16.h>
#include <stdint.h>

// CDNA5 (gfx1250) wave32 WMMA: fp32 16x16x4 (the only f32 WMMA shape).
typedef float v2f __attribute__((ext_vector_type(2)));
typedef float v8f __attribute__((ext_vector_type(8)));

static __device__ __forceinline__ v8f wmma_f32_16x16x4(v2f a, v2f b, v8f c) {
  // (neg_a, A, neg_b, B, c_mod, C, reuse_a, reuse_b)
  return __builtin_amdgcn_wmma_f32_16x16x4_f32(false, a, false, b, (short)0, c,
                                               false, false);
}

// ---------------------------------------------------------------------------
// f1[b][o][n] = sum_{c<3} w0[o][c]*x[b][c][n] + b0[o]     (O=8 fixed)
// ---------------------------------------------------------------------------
__global__ void k_lin3(const float* __restrict__ x, const float* __restrict__ w0,
                       const float* __restrict__ b0, float* __restrict__ f1,
                       int B, int N) {
  int t = blockIdx.x * blockDim.x + threadIdx.x;
  int total = B * 8 * N;
  if (t >= total) return;
  int n = t % N;
  int o = (t / N) % 8;
  int b = t / (N * 8);
  const float* xb = x + (size_t)b * 3 * N;
  float acc = b0[o];
  acc += w0[o * 3 + 0] * xb[n];
  acc += w0[o * 3 + 1] * xb[N + n];
  acc += w0[o * 3 + 2] * xb[2 * N + n];
  f1[t] = acc;
}

// ---------------------------------------------------------------------------
// kNN (k=16). One wave (block of 32) handles 32 queries. Per 16-key tile:
// two V_WMMA_F32_16X16X4_F32 compute the 32x16 query.key dot tile (K=3 padded
// to 4 with zeros), scores s = |k|^2 - 2*dot (row-constant |q|^2 dropped:
// ordering-invariant). Tile transposed through LDS (row stride 20 floats ->
// 16B-aligned rows, conflict-free writes, <=2-way-conflicted b128 reads),
// each lane bulk-loads its 16 scores with 4x ds_load_b128 and keeps a sorted
// top-16 (dist,idx) list in registers (stable insert: ties keep the earlier
// key index, matching jax.lax.top_k).
// ---------------------------------------------------------------------------
__global__ void k_knn(const float* __restrict__ cq, int Nq,
                      const float* __restrict__ ck, int Nk,
                      int* __restrict__ idxout, int B) {
  __shared__ alignas(16) float s[32 * 20];
  const int lane = threadIdx.x;
  const int qtiles = Nq >> 5;
  const int b = blockIdx.x / qtiles;
  const int qt = blockIdx.x % qtiles;
  if (b >= B) return;
  const int q0 = qt * 32;
  const float* cqb = cq + (size_t)b * 3 * Nq;
  const float* ckb = ck + (size_t)b * 3 * Nk;
  const bool hi = lane >= 16;
  const int ql = lane & 15;

  // A operands: lanes 0-15 hold (x,y) of their query (K=0,1); lanes 16-31
  // hold (z,0) (K=2,3).  a0 -> queries q0..q0+15, a1 -> q0+16..q0+31.
  v2f a0, a1;
  {
    int q = q0 + ql;
    a0.x = hi ? cqb[2 * Nq + q] : cqb[q];
    a0.y = hi ? 0.0f : cqb[Nq + q];
    a1.x = hi ? cqb[2 * Nq + q + 16] : cqb[q + 16];
    a1.y = hi ? 0.0f : cqb[Nq + q + 16];
  }

  float bd[16];
  int bi[16];
#pragma unroll
  for (int t = 0; t < 16; ++t) { bd[t] = 3.0e38f; bi[t] = 0; }

  for (int k0 = 0; k0 < Nk; k0 += 16) {
    int key = k0 + ql;
    // Prefetch next key tile (uniform branch; stream is fully predictable).
    if (k0 + 16 < Nk) {
      __builtin_prefetch(ckb + key + 16, 0, 1);
      __builtin_prefetch(ckb + Nk + key + 16, 0, 1);
      __builtin_prefetch(ckb + 2 * Nk + key + 16, 0, 1);
    }
    float kx = ckb[key], ky = ckb[Nk + key], kz = ckb[2 * Nk + key];
    v2f bm;
    bm.x = hi ? kz : kx;
    bm.y = hi ? 0.0f : ky;
    float kk = kx * kx + ky * ky + kz * kz;
    v8f d0 = {0, 0, 0, 0, 0, 0, 0, 0};
    v8f d1 = {0, 0, 0, 0, 0, 0, 0, 0};
    d0 = wmma_f32_16x16x4(a0, bm, d0);
    d1 = wmma_f32_16x16x4(a1, bm, d1);
    // D layout: vgpr r, lane l -> row r + 8*(l>=16), col l&15.
    int mb = hi ? 8 : 0;
#pragma unroll
    for (int r = 0; r < 8; ++r) {
      s[(mb + r) * 20 + ql] = kk - 2.0f * d0[r];
      s[(16 + mb + r) * 20 + ql] = kk - 2.0f * d1[r];
    }
    __syncthreads();
    // Bulk-read this lane's row (its query's 16 candidate scores).
    const float4* rowp = (const float4*)(&s[lane * 20]);
    float4 c0 = rowp[0], c1 = rowp[1], c2 = rowp[2], c3 = rowp[3];
    float sc[16] = {c0.x, c0.y, c0.z, c0.w, c1.x, c1.y, c1.z, c1.w,
                    c2.x, c2.y, c2.z, c2.w, c3.x, c3.y, c3.z, c3.w};
    float worst = bd[15];
#pragma unroll
    for (int nn = 0; nn < 16; ++nn) {
      if (sc[nn] < worst) {
        float dt = sc[nn];
        int it = k0 + nn;
#pragma unroll
        for (int t = 0; t < 16; ++t) {  // stable bubble-insert
          bool sm = dt < bd[t];
          float od = bd[t];
          int oi = bi[t];
          bd[t] = sm ? dt : od;
          bi[t] = sm ? it : oi;
          dt = sm ? od : dt;
          it = sm ? oi : it;
        }
        worst = bd[15];
      }
    }
    __syncthreads();
  }
  int* outp = idxout + ((size_t)b * Nq + q0 + lane) * 16;
#pragma unroll
  for (int t = 0; t < 16; ++t) outp[t] = bi[t];
}

// ---------------------------------------------------------------------------
// z[b][o][n] = sum_c (w[o][C+c]-w[o][c]) * xq[b][c][n]
// (the broadcast half of the edge feature, folded algebraically)
// ---------------------------------------------------------------------------
__global__ void k_zproj(const float* __restrict__ xq, int Nq,
                        const float* __restrict__ w, int C, int O,
                        float* __restrict__ z, int B) {
  int t = blockIdx.x * blockDim.x + threadIdx.x;
  int total = B * O * Nq;
  if (t >= total) return;
  int n = t % Nq;
  int o = (t / Nq) % O;
  int b = t / (Nq * O);
  const float* wr = w + (size_t)o * 2 * C;
  const float* xb = xq + (size_t)b * C * Nq + n;
  float acc = 0.0f;
  for (int c = 0; c < C; ++c) acc += (wr[C + c] - wr[c]) * xb[(size_t)c * Nq];
  z[t] = acc;
}

// ---------------------------------------------------------------------------
// Edge-conv GEMM with fused gather: y[b][o][n][j] =
//   sum_c w[o][c] * xk[b][c][ idx[b][n][j] ]  +  z[b][o][n]
// One wave per (b, 16-row tile of O, query n): the 16 columns of the WMMA
// tile are exactly query n's 16 neighbors. K = C chained in steps of 4 via
// V_WMMA_F32_16X16X4_F32 (C/4 is even for every layer -> unroll 2).
// ---------------------------------------------------------------------------
__global__ void k_edgeconv(const float* __restrict__ xk, int Nk,
                           const int* __restrict__ idx,
                           const float* __restrict__ w,
                           const float* __restrict__ z, float* __restrict__ y,
                           int B, int C, int O, int Nq) {
  const int lane = threadIdx.x & 31;
  const int wave = blockIdx.x * (blockDim.x >> 5) + (threadIdx.x >> 5);
  const int MT = O >> 4;
  int n = wave % Nq;
  int mt = (wave / Nq) % MT;
  int b = wave / (Nq * MT);
  if (b >= B) return;
  const int col = lane & 15;
  const bool hi = lane >= 16;
  const int key = idx[((size_t)b * Nq + n) * 16 + col];
  const float* xkb = xk + (size_t)b * C * Nk + key;
  const float* wr = w + (size_t)(mt * 16 + col) * 2 * C;  // A row = lane&15
  const int kh = hi ? 2 : 0;
  v8f acc = {0, 0, 0, 0, 0, 0, 0, 0};
#pragma unroll 2
  for (int k0 = 0; k0 < C; k0 += 4) {
    v2f a, bm;
    a.x = wr[k0 + kh];
    a.y = wr[k0 + kh + 1];
    bm.x = xkb[(size_t)(k0 + kh) * Nk];
    bm.y = xkb[(size_t)(k0 + kh + 1) * Nk];
    acc = wmma_f32_16x16x4(a, bm, acc);
  }
  const float* zb = z + ((size_t)b * O + mt * 16) * Nq + n;
  float* yb = y + (((size_t)b * O + mt * 16) * Nq + n) * 16;
  const int mb = hi ? 8 : 0;
#pragma unroll
  for (int r = 0; r < 8; ++r) {
    int m = mb + r;
    yb[(size_t)m * Nq * 16 + col] = acc[r] + zb[(size_t)m * Nq];
  }
}

// ---------------------------------------------------------------------------
// GroupNorm stats: one 256-thread block per (b, group). groups = 4.
// float4 streaming (group span is contiguous & 16B aligned).
// stats[(b*4+g)*2] = mean, [+1] = rsqrt(var + eps)  (biased var, like jnp.var)
// ---------------------------------------------------------------------------
__global__ void k_gnstats(const float* __restrict__ y, int O, int Nq,
                          float* __restrict__ stats, int B) {
  __shared__ float sr[256], sr2[256];
  int g = blockIdx.x & 3;
  int b = blockIdx.x >> 2;
  int Cg = O >> 2;
  size_t base = ((size_t)b * O + (size_t)g * Cg) * Nq * 16;
  size_t cnt = (size_t)Cg * Nq * 16;
  const float4* yp = (const float4*)(y + base);
  size_t cnt4 = cnt >> 2;
  float s = 0.0f, ss = 0.0f;
  for (size_t i = threadIdx.x; i < cnt4; i += 256) {
    float4 v = yp[i];
    s += v.x + v.y + v.z + v.w;
    ss += v.x * v.x + v.y * v.y + v.z * v.z + v.w * v.w;
  }
  sr[threadIdx.x] = s;
  sr2[threadIdx.x] = ss;
  __syncthreads();
  for (int st = 128; st > 0; st >>= 1) {
    if ((int)threadIdx.x < st) {
      sr[threadIdx.x] += sr[threadIdx.x + st];
      sr2[threadIdx.x] += sr2[threadIdx.x + st];
    }
    __syncthreads();
  }
  if (threadIdx.x == 0) {
    float inv = 1.0f / (float)cnt;
    float mu = sr[0] * inv;
    float var = sr2[0] * inv - mu * mu;
    stats[(b * 4 + g) * 2] = mu;
    stats[(b * 4 + g) * 2 + 1] = rsqrtf(var + 1e-5f);
  }
}

// ---------------------------------------------------------------------------
// GN apply + LeakyReLU(0.2) + max over the 16 neighbors. Thread per (b,o,n);
// the 16 neighbor values are one contiguous 64B-aligned run -> 4x b128 loads.
// ---------------------------------------------------------------------------
__global__ void k_gnmax(const float* __restrict__ y,
                        const float* __restrict__ stats,
                        const float* __restrict__ gamma,
                        const float* __restrict__ beta, int O, int Nq,
                        float* __restrict__ f, int B) {
  int t = blockIdx.x * blockDim.x + threadIdx.x;
  int total = B * O * Nq;
  if (t >= total) return;
  int o = (t / Nq) % O;
  int b = t / (Nq * O);
  int g = o / (O >> 2);
  float mu = stats[(b * 4 + g) * 2];
  float rs = stats[(b * 4 + g) * 2 + 1];
  float gm = gamma[o] * rs;
  float bt = beta[o] - mu * gm;
  const float4* yp = (const float4*)(y + (size_t)t * 16);
  float m = -3.0e38f;
#pragma unroll
  for (int j = 0; j < 4; ++j) {
    float4 v4 = yp[j];
    float vv[4] = {v4.x, v4.y, v4.z, v4.w};
#pragma unroll
    for (int e = 0; e < 4; ++e) {
      float v = vv[e] * gm + bt;
      v = v > 0.0f ? v : 0.2f * v;
      m = fmaxf(m, v);
    }
  }
  f[t] = m;
}

// ---------------------------------------------------------------------------
// Farthest-point sampling: one 256-thread block per batch, points + running
// min-distances resident in registers (PER points/thread), sequential n steps
// with LDS argmax reduction (first-index tie-break, like jnp.argmax).
// Emits the incoming 'far' each step (far0 = 0), matching the jax.lax.scan.
// ---------------------------------------------------------------------------
template <int PER>
__global__ void k_fps(const float* __restrict__ coor, int N, int n,
                      int* __restrict__ out) {
  __shared__ float rd[256];
  __shared__ int ri[256];
  __shared__ float cent[3];
  const int b = blockIdx.x;
  const int tid = threadIdx.x;
  const float* cb = coor + (size_t)b * 3 * N;
  float px[PER], py[PER], pz[PER], dd[PER];
#pragma unroll
  for (int i = 0; i < PER; ++i) {
    int p = i * 256 + tid;
    px[i] = cb[p];
    py[i] = cb[N + p];
    pz[i] = cb[2 * N + p];
    dd[i] = 1.0e10f;
  }
  int far = 0;
  for (int it = 0; it < n; ++it) {
    if (tid == 0) {
      out[(size_t)b * n + it] = far;
      cent[0] = cb[far];
      cent[1] = cb[N + far];
      cent[2] = cb[2 * N + far];
    }
    __syncthreads();
    float cx = cent[0], cy = cent[1], cz = cent[2];
    float bestd = -1.0f;
    int besti = 0;
#pragma unroll
    for (int i = 0; i < PER; ++i) {
      float dx = px[i] - cx, dy = py[i] - cy, dz = pz[i] - cz;
      float d = dx * dx + dy * dy + dz * dz;
      float nd = fminf(dd[i], d);
      dd[i] = nd;
      if (nd > bestd) {
        bestd = nd;
        besti = i * 256 + tid;
      }
    }
    rd[tid] = bestd;
    ri[tid] = besti;
    __syncthreads();
    for (int st = 128; st > 0; st >>= 1) {
      if (tid < st) {
        float od = rd[tid + st];
        int oi = ri[tid + st];
        if (od > rd[tid] || (od == rd[tid] && oi < ri[tid])) {
          rd[tid] = od;
          ri[tid] = oi;
        }
      }
      __syncthreads();
    }
    far = ri[0];
    __syncthreads();
  }
}

// ---------------------------------------------------------------------------
// dst[b][c][i] = src[b][c][ fidx[b][i] ]
// ---------------------------------------------------------------------------
__global__ void k_gather_ch(const float* __restrict__ src, int C, int Nsrc,
                            const int* __restrict__ fidx, int n,
                            float* __restrict__ dst, int B) {
  int t = blockIdx.x * blockDim.x + threadIdx.x;
  int total = B * C * n;
  if (t >= total) return;
  int i = t % n;
  int c = (t / n) % C;
  int b = t / (n * C);
  int p = fidx[(size_t)b * n + i];
  dst[t] = src[((size_t)b * C + c) * Nsrc + p];
}

// ===========================================================================
extern "C" void kernel_launch(void* const* d_in, const int* in_sizes, int n_in,
                              void* d_out, int out_size, void* d_ws,
                              size_t ws_size, hipStream_t stream) {
  (void)in_sizes; (void)n_in; (void)out_size; (void)ws_size;
  const float* x = (const float*)d_in[0];
  const float* w0 = (const float*)d_in[1];
  const float* b0 = (const float*)d_in[2];
  const float* w1 = (const float*)d_in[3];
  const float* g1 = (const float*)d_in[4];
  const float* bg1 = (const float*)d_in[5];
  const float* w2 = (const float*)d_in[6];
  const float* g2 = (const float*)d_in[7];
  const float* bg2 = (const float*)d_in[8];
  const float* w3 = (const float*)d_in[9];
  const float* g3 = (const float*)d_in[10];
  const float* bg3 = (const float*)d_in[11];
  const float* w4 = (const float*)d_in[12];
  const float* g4 = (const float*)d_in[13];
  const float* bg4 = (const float*)d_in[14];
  float* out = (float*)d_out;

  const int B = 8, N0 = 4096, N1 = 512, N2 = 128;

  // bump-allocate workspace (~82 MB total)
  char* ws = (char*)d_ws;
  size_t off = 0;
  auto alloc = [&](size_t bytes) -> void* {
    void* p = ws + off;
    off = (off + bytes + 255) & ~(size_t)255;
    return p;
  };
  float* F1 = (float*)alloc((size_t)B * 8 * N0 * 4);        // (8,8,4096)
  int* IDX = (int*)alloc((size_t)B * N0 * 16 * 4);          // knn idx (reused)
  float* Z = (float*)alloc((size_t)B * 32 * N0 * 4);        // z proj (reused)
  float* Y = (float*)alloc((size_t)B * 32 * N0 * 16 * 4);   // conv out (reused)
  float* F2 = (float*)alloc((size_t)B * 32 * N0 * 4);       // (8,32,4096)
  float* ST = (float*)alloc(256);                           // GN stats
  int* FI = (int*)alloc((size_t)B * N1 * 4);                // fps idx (reused)
  float* CQ1 = (float*)alloc((size_t)B * 3 * N1 * 4);       // coor @512
  float* FQ1 = (float*)alloc((size_t)B * 32 * N1 * 4);      // f_q @512
  float* F3 = (float*)alloc((size_t)B * 64 * N1 * 4);       // (8,64,512)
  float* F4 = (float*)alloc((size_t)B * 64 * N1 * 4);       // (8,64,512)
  float* FQ2 = (float*)alloc((size_t)B * 64 * N2 * 4);      // f_q @128

  // f1 = w0 @ x + b0
  k_lin3<<<(B * 8 * N0 + 255) / 256, 256, 0, stream>>>(x, w0, b0, F1, B, N0);

  // ---- layer 1: knn(4096,4096), edgeconv C=8 -> O=32, GN, max ----
  k_knn<<<B * (N0 / 32), 32, 0, stream>>>(x, N0, x, N0, IDX, B);
  k_zproj<<<(B * 32 * N0 + 255) / 256, 256, 0, stream>>>(F1, N0, w1, 8, 32, Z, B);
  k_edgeconv<<<B * 2 * N0 / 4, 128, 0, stream>>>(F1, N0, IDX, w1, Z, Y, B, 8, 32, N0);
  k_gnstats<<<B * 4, 256, 0, stream>>>(Y, 32, N0, ST, B);
  k_gnmax<<<(B * 32 * N0 + 255) / 256, 256, 0, stream>>>(Y, ST, g1, bg1, 32, N0, F2, B);

  // ---- FPS 4096 -> 512, gather coor + f2 ----
  k_fps<16><<<B, 256, 0, stream>>>(x, N0, N1, FI);
  k_gather_ch<<<(B * 3 * N1 + 255) / 256, 256, 0, stream>>>(x, 3, N0, FI, N1, CQ1, B);
  k_gather_ch<<<(B * 32 * N1 + 255) / 256, 256, 0, stream>>>(F2, 32, N0, FI, N1, FQ1, B);

  // ---- layer 2: knn(q=512, k=4096), edgeconv C=32 -> O=64 ----
  k_knn<<<B * (N1 / 32), 32, 0, stream>>>(CQ1, N1, x, N0, IDX, B);
  k_zproj<<<(B * 64 * N1 + 255) / 256, 256, 0, stream>>>(FQ1, N1, w2, 32, 64, Z, B);
  k_edgeconv<<<B * 4 * N1 / 4, 128, 0, stream>>>(F2, N0, IDX, w2, Z, Y, B, 32, 64, N1);
  k_gnstats<<<B * 4, 256, 0, stream>>>(Y, 64, N1, ST, B);
  k_gnmax<<<(B * 64 * N1 + 255) / 256, 256, 0, stream>>>(Y, ST, g2, bg2, 64, N1, F3, B);

  // ---- layer 3: knn(512,512), edgeconv C=64 -> O=64 ----
  k_knn<<<B * (N1 / 32), 32, 0, stream>>>(CQ1, N1, CQ1, N1, IDX, B);
  k_zproj<<<(B * 64 * N1 + 255) / 256, 256, 0, stream>>>(F3, N1, w3, 64, 64, Z, B);
  k_edgeconv<<<B * 4 * N1 / 4, 128, 0, stream>>>(F3, N1, IDX, w3, Z, Y, B, 64, 64, N1);
  k_gnstats<<<B * 4, 256, 0, stream>>>(Y, 64, N1, ST, B);
  k_gnmax<<<(B * 64 * N1 + 255) / 256, 256, 0, stream>>>(Y, ST, g3, bg3, 64, N1, F4, B);

  // ---- FPS 512 -> 128; coor_q2 straight into d_out ----
  k_fps<2><<<B, 256, 0, stream>>>(CQ1, N1, N2, FI);
  k_gather_ch<<<(B * 3 * N2 + 255) / 256, 256, 0, stream>>>(CQ1, 3, N1, FI, N2, out, B);
  k_gather_ch<<<(B * 64 * N2 + 255) / 256, 256, 0, stream>>>(F4, 64, N1, FI, N2, FQ2, B);

  // ---- layer 4: knn(q=128, k=512), edgeconv C=64 -> O=128, f5 -> d_out ----
  k_knn<<<B * (N2 / 32), 32, 0, stream>>>(out, N2, CQ1, N1, IDX, B);
  k_zproj<<<(B * 128 * N2 + 255) / 256, 256, 0, stream>>>(FQ2, N2, w4, 64, 128, Z, B);
  k_edgeconv<<<B * 8 * N2 / 4, 128, 0, stream>>>(F4, N1, IDX, w4, Z, Y, B, 64, 128, N2);
  k_gnstats<<<B * 4, 256, 0, stream>>>(Y, 128, N2, ST, B);
  k_gnmax<<<(B * 128 * N2 + 255) / 256, 256, 0, stream>>>(Y, ST, g4, bg4, 128, N2,
                                                          out + 3072, B);
}